// RWaveCenteredPositionalEncoding_44023414784139
// MI455X (gfx1250) — compile-verified
//
#include <hip/hip_runtime.h>
#include <math.h>

// Problem constants (match reference): B=32, S=8192, D=256, P=64, MAX_LEN=8192
#define BB 32
#define SS 8192
#define DD 256
#define PP 64
#define WAVES_PER_BLOCK 8   // one wave32 per sequence position

#define PI_F    3.14159265358979323846f
#define INV_2PI 0.15915494309189535f
// div[k] = 10000^(-k/128) = exp2(-k * log2(10000)/128); log2(10000)/128 = 0.103810252965223
#define L2_DIV  0.10381025296522300f

#define HAVE_TDM (__has_builtin(__builtin_amdgcn_tensor_load_to_lds) && \
                  __has_builtin(__builtin_amdgcn_s_wait_tensorcnt))

typedef unsigned int u32x4 __attribute__((ext_vector_type(4)));
typedef int          i32x4 __attribute__((ext_vector_type(4)));
typedef int          i32x8 __attribute__((ext_vector_type(8)));
typedef float        f32x4 __attribute__((ext_vector_type(4)));

__global__ __launch_bounds__(256)
void rwave_pe_kernel(const float* __restrict__ pe,        // [MAX_LEN, D]
                     const int*   __restrict__ peaks_g,   // [B, P] sorted
                     const int*   __restrict__ n_peaks,   // [B]
                     float*       __restrict__ out)       // [B, S, D]
{
    __shared__ __align__(16) int   s_pk[PP];
    __shared__ __align__(16) float s_rr[PP];
    __shared__ __align__(16) float s_div[DD / 2];  // div[k] * (1/2pi), for hw v_sin/v_cos
    __shared__ float s_avg;

    const int b   = blockIdx.y;
    const int tid = threadIdx.x;
    const int n   = n_peaks[b];

    // ---- block prologue: peaks -> LDS via Tensor Data Mover (64x1 tile of 4B) ----
#if HAVE_TDM
    if (tid < 32) {  // wave 0 issues the TDM descriptor (uniform operands)
        // Flat LDS pointers carry the LDS byte offset in the low 32 bits (ISA 10.2 aperture rule).
        const unsigned lds_off = (unsigned)(size_t)(const void*)s_pk;
        const unsigned long long ga =
            (unsigned long long)(size_t)(const void*)(peaks_g + (size_t)b * PP);

        u32x4 g0;
        g0[0] = 1u;                                          // count=1 (valid), no gather
        g0[1] = lds_off;                                     // lds_addr (bytes)
        g0[2] = (unsigned)(ga & 0xFFFFFFFFu);                // global_addr[31:0]
        g0[3] = (unsigned)((ga >> 32) & 0x01FFFFFFu)         // global_addr[56:32]
                | (2u << 30);                                // type = 2 ("image")
        i32x8 g1;
        g1[0] = (int)(2u << 16);                             // workgroup_mask=0, data_size=2 (4B)
        g1[1] = (int)((unsigned)PP << 16);                   // tensor_dim0 = 64 (lo16)
        g1[2] = (int)(1u << 16);                             // tensor_dim0 hi=0, tensor_dim1 = 1
        g1[3] = (int)((unsigned)PP << 16);                   // tensor_dim1 hi=0, tile_dim0 = 64
        g1[4] = 1;                                           // tile_dim1 = 1, tile_dim2 = 0
        g1[5] = PP;                                          // tensor_dim0_stride = 64 (lo32)
        g1[6] = 0;                                           // stride hi, tensor_dim1_stride lo
        g1[7] = 0;
        i32x4 gz; gz[0] = 0; gz[1] = 0; gz[2] = 0; gz[3] = 0; // groups 2/3 unused (2D tensor)
#if __clang_major__ >= 23
        i32x8 gz8; gz8[0]=0; gz8[1]=0; gz8[2]=0; gz8[3]=0; gz8[4]=0; gz8[5]=0; gz8[6]=0; gz8[7]=0;
        __builtin_amdgcn_tensor_load_to_lds(g0, g1, gz, gz, gz8, 0);
#else
        __builtin_amdgcn_tensor_load_to_lds(g0, g1, gz, gz, 0);
#endif
        __builtin_amdgcn_s_wait_tensorcnt(0);
    }
#else
    if (tid < PP) {
        s_pk[tid] = peaks_g[b * PP + tid];
    }
#endif
    if (tid >= 128 && tid < 128 + DD / 2) {
        const int k = tid - 128;
        s_div[k] = exp2f(-L2_DIV * (float)k) * INV_2PI;
    }
    __syncthreads();

    // rr[0] = pk[1]-pk[0]; rr[i] = pk[i]-pk[i-1]; n==1 -> MAX_LEN/2
    if (tid < PP) {
        float r;
        if (n == 1) {
            r = (float)(SS / 2);
        } else {
            const int i = (tid == 0) ? 1 : tid;
            r = (float)(s_pk[i] - s_pk[i - 1]);
        }
        s_rr[tid] = r;
    }
    if (tid == PP) {
        // sum_{i<n} rr[i] telescopes to rr[0] + (pk[n-1]-pk[0])
        float avg = 1.0f;
        if (n == 1) {
            avg = (float)(SS / 2);
        } else if (n >= 2) {
            const float sum = (float)((s_pk[1] - s_pk[0]) + (s_pk[n - 1] - s_pk[0]));
            avg = fmaxf(1.0f, sum / (float)n);
        }
        s_avg = avg;
    }
    __syncthreads();

    const int wave = tid >> 5;
    const int lane = tid & 31;
    const int s    = blockIdx.x * WAVES_PER_BLOCK + wave;
    float* rowp = out + ((size_t)b * SS + (size_t)s) * DD;

    if (n <= 0) {
        // empty-peak fallback: copy pe[s, :]
        const f32x4* per = (const f32x4*)(pe + (size_t)s * DD);
        f32x4 a = per[lane * 2];
        f32x4 c = per[lane * 2 + 1];
        __builtin_nontemporal_store(a, (f32x4*)rowp + lane * 2);
        __builtin_nontemporal_store(c, (f32x4*)rowp + lane * 2 + 1);
        return;
    }

    // ---- nearest valid peak, argmin-first semantics (wave-uniform, pure int) ----
    int lo = 0, hi = n;
    while (lo < hi) {
        const int mid = (lo + hi) >> 1;
        if (s_pk[mid] < s) lo = mid + 1; else hi = mid;
    }
    int ni;
    if (lo >= n) {
        ni = n - 1;
    } else if (lo == 0) {
        ni = 0;
    } else {
        const int dl = s - s_pk[lo - 1];
        const int dr = s_pk[lo] - s;
        ni = (dl <= dr) ? (lo - 1) : lo;
    }
    // argmin returns first occurrence: walk left over duplicate peak values
    while (ni > 0 && s_pk[ni - 1] == s_pk[ni]) --ni;

    const int   npos_i = s_pk[ni];
    const float nrr    = fmaxf(s_rr[ni], 1.0f);
    const float rel    = (float)(s - npos_i) / nrr;
    const float arel   = fabsf(rel);
    const float phase  = arel - floorf(arel);   // |rel| mod 1.0

    float v0, v1, v2, v3, v4, v5, v6, v7;
    if (lane == 0) {
        // special channels 0..7
        const float avg = s_avg;
        v0 = sinf(rel * PI_F);                          // large arg range -> libm
        v1 = cosf(rel * PI_F);
        v2 = __builtin_amdgcn_sinf(phase);              // sin(2*pi*phase), phase in [0,1)
        v3 = __builtin_amdgcn_cosf(phase);
        v4 = fminf(arel, 1.0f);                         // |pos-npos|/nrr clipped
        v5 = (npos_i == s) ? 1.0f : 0.0f;               // is_peak
        v6 = fminf(fmaxf(nrr / avg, 0.5f), 2.0f);       // norm_rr
        const float prr = s_rr[(ni > 0) ? (ni - 1) : 0];
        v7 = (ni > 0) ? fminf(fabsf(nrr - prr) / avg, 1.0f) : 0.0f;  // hrv
    } else {
        // channels 8*lane .. 8*lane+7 : pairs k = 4*lane .. 4*lane+3
        const f32x4 dv = ((const f32x4*)s_div)[lane];   // pre-scaled by 1/2pi
        const float a0 = phase * dv.x;
        const float a1 = phase * dv.y;
        const float a2 = phase * dv.z;
        const float a3 = phase * dv.w;
        v0 = __builtin_amdgcn_sinf(a0);  v1 = __builtin_amdgcn_cosf(a0);
        v2 = __builtin_amdgcn_sinf(a1);  v3 = __builtin_amdgcn_cosf(a1);
        v4 = __builtin_amdgcn_sinf(a2);  v5 = __builtin_amdgcn_cosf(a2);
        v6 = __builtin_amdgcn_sinf(a3);  v7 = __builtin_amdgcn_cosf(a3);
    }

    f32x4 o0; o0.x = v0; o0.y = v1; o0.z = v2; o0.w = v3;
    f32x4 o1; o1.x = v4; o1.y = v5; o1.z = v6; o1.w = v7;
    // streaming (non-temporal) coalesced b128 stores: 1KB per wave per position
    __builtin_nontemporal_store(o0, (f32x4*)rowp + lane * 2);
    __builtin_nontemporal_store(o1, (f32x4*)rowp + lane * 2 + 1);
}

extern "C" void kernel_launch(void* const* d_in, const int* in_sizes, int n_in,
                              void* d_out, int out_size, void* d_ws, size_t ws_size,
                              hipStream_t stream) {
    // inputs (setup_inputs order): x [B,S,1] f32 (unused), pe [MAX_LEN,D] f32,
    // r_peak_indices [B,P] i32, n_peaks [B] i32
    const float* pe    = (const float*)d_in[1];
    const int*   peaks = (const int*)d_in[2];
    const int*   npk   = (const int*)d_in[3];
    float*       out   = (float*)d_out;

    dim3 grid(SS / WAVES_PER_BLOCK, BB);
    dim3 block(32 * WAVES_PER_BLOCK);
    rwave_pe_kernel<<<grid, block, 0, stream>>>(pe, peaks, npk, out);
}